// LConvBilin_26199300505664
// MI455X (gfx1250) — compile-verified
//
#include <hip/hip_runtime.h>

// ---------------------------------------------------------------------------
// LConv bilinear layer, refactored as:
//   out_w[u] = sum_v w2[v] @ M[u,v],  M[u,v] = sum_w weight[u,v,w] * t_w2[w]
// M is one big GEMM:  M[48(pad), S*18] = W[36->48, 41->44] x Tw2[44, S*18]
// executed with v_wmma_f32_16x16x4_f32 (f32 matches reference precision).
// M rows are padded to 48 so the WMMA C-tile store is branch-free (EXEC
// stays all-ones for the whole stage-B wave).
// ---------------------------------------------------------------------------

#define NSITE 16384              // 16*16*8*8
#define SITE_STRIDE 144          // 8 matrices * 9 elems * 2 (re,im)
#define NCOLS (NSITE * 18)       // 294912 GEMM columns (3*3*2 per site)
#define KPAD 44                  // 41 -> 44 (11 k-steps of 4)
#define MPAD 48                  // 36 -> 48 (3 m-tiles of 16)
#define MROWS 36
#define NTILES (NCOLS / 16)      // 18432

typedef __attribute__((ext_vector_type(2))) float v2f;
typedef __attribute__((ext_vector_type(8))) float v8f;

struct C33 { float re[9]; float im[9]; };

__device__ __forceinline__ void loadC33(const float* __restrict__ p, C33& A) {
#pragma unroll
  for (int e = 0; e < 9; ++e) { A.re[e] = p[2 * e]; A.im[e] = p[2 * e + 1]; }
}

__device__ __forceinline__ void storeC33(float* __restrict__ p, const C33& A) {
#pragma unroll
  for (int e = 0; e < 9; ++e) { p[2 * e] = A.re[e]; p[2 * e + 1] = A.im[e]; }
}

// store conj-transpose (dagger) of A
__device__ __forceinline__ void storeC33dag(float* __restrict__ p, const C33& A) {
#pragma unroll
  for (int j = 0; j < 3; ++j)
#pragma unroll
    for (int k = 0; k < 3; ++k) {
      p[(j * 3 + k) * 2]     =  A.re[k * 3 + j];
      p[(j * 3 + k) * 2 + 1] = -A.im[k * 3 + j];
    }
}

// D = A @ B (complex 3x3)
__device__ __forceinline__ void cmul33(const C33& A, const C33& B, C33& D) {
#pragma unroll
  for (int i = 0; i < 3; ++i)
#pragma unroll
    for (int k = 0; k < 3; ++k) {
      float r = 0.f, m = 0.f;
#pragma unroll
      for (int j = 0; j < 3; ++j) {
        float ar = A.re[i * 3 + j], ai = A.im[i * 3 + j];
        float br = B.re[j * 3 + k], bi = B.im[j * 3 + k];
        r = fmaf(ar, br, fmaf(-ai, bi, r));
        m = fmaf(ar, bi, fmaf(ai, br, m));
      }
      D.re[i * 3 + k] = r; D.im[i * 3 + k] = m;
    }
}

// D = A @ B-dagger (complex 3x3)
__device__ __forceinline__ void cmul33_bdag(const C33& A, const C33& B, C33& D) {
#pragma unroll
  for (int i = 0; i < 3; ++i)
#pragma unroll
    for (int k = 0; k < 3; ++k) {
      float r = 0.f, m = 0.f;
#pragma unroll
      for (int j = 0; j < 3; ++j) {
        float ar = A.re[i * 3 + j], ai = A.im[i * 3 + j];
        float br = B.re[k * 3 + j], bi = -B.im[k * 3 + j];
        r = fmaf(ar, br, fmaf(-ai, bi, r));
        m = fmaf(ar, bi, fmaf(ai, br, m));
      }
      D.re[i * 3 + k] = r; D.im[i * 3 + k] = m;
    }
}

// ---------------------------------------------------------------------------
// Kernel 0: pad weight (4,9,41) -> Aw[48][44], zero padded.
// ---------------------------------------------------------------------------
__global__ void lconv_initAw(const float* __restrict__ weight, float* __restrict__ Aw) {
  int tid = blockIdx.x * blockDim.x + threadIdx.x;
  if (tid >= MPAD * KPAD) return;
  int r = tid / KPAD, k = tid % KPAD;
  float v = 0.f;
  if (r < MROWS && k < 41) v = weight[r * 41 + k];  // weight[(u*9+v)*41 + w]
  Aw[tid] = v;
}

// ---------------------------------------------------------------------------
// Kernel 1: per (site, m): copy u->out; build t_w2 rows of Tw2[44][NCOLS]:
//   row m           = w[m]
//   row 4+a*4+m     = U_a(x) w[m](x+a) U_a(x)^dag
//   row 20+r        = row r dagger  (r in 0..19)
//   row 40          = identity;  rows 41..43 = 0
// ---------------------------------------------------------------------------
__global__ void lconv_stageA(const float* __restrict__ x, float* __restrict__ T,
                             float* __restrict__ out) {
  int tid = blockIdx.x * blockDim.x + threadIdx.x;   // 0 .. 4*NSITE-1
  int s = tid >> 2;
  int m = tid & 3;
  if (s >= NSITE) return;
  const float* xs = x + (size_t)s * SITE_STRIDE;

  // pass-through of gauge links: out[:, :, 0:4] = u
  {
    const float* up = xs + m * 18;
    float* op = out + (size_t)s * SITE_STRIDE + m * 18;
#pragma unroll
    for (int e = 0; e < 18; ++e) op[e] = up[e];
  }

  C33 w; loadC33(xs + (4 + m) * 18, w);
  storeC33(T + (size_t)m * NCOLS + s * 18, w);
  storeC33dag(T + (size_t)(20 + m) * NCOLS + s * 18, w);

  // site coords: dims (16,16,8,8), strides (1024,64,8,1)
  int d3 = s & 7, d2 = (s >> 3) & 7, d1 = (s >> 6) & 15, d0 = (s >> 10) & 15;

#pragma unroll
  for (int a = 0; a < 4; ++a) {
    int c0 = d0, c1 = d1, c2 = d2, c3 = d3;
    if (a == 0) c0 = (c0 + 1) & 15;
    else if (a == 1) c1 = (c1 + 1) & 15;
    else if (a == 2) c2 = (c2 + 1) & 7;
    else c3 = (c3 + 1) & 7;
    int sn = ((c0 * 16 + c1) * 8 + c2) * 8 + c3;

    C33 U;  loadC33(xs + a * 18, U);
    C33 wn; loadC33(x + (size_t)sn * SITE_STRIDE + (4 + m) * 18, wn);
    C33 t1; cmul33(U, wn, t1);        // U * w(x+a)
    C33 t2; cmul33_bdag(t1, U, t2);   // ... * U^dag

    int r = 4 + a * 4 + m;
    storeC33(T + (size_t)r * NCOLS + s * 18, t2);
    storeC33dag(T + (size_t)(20 + r) * NCOLS + s * 18, t2);
  }

  if (m == 0) {
    float* p = T + (size_t)40 * NCOLS + s * 18;     // identity row (complex I)
#pragma unroll
    for (int j = 0; j < 3; ++j)
#pragma unroll
      for (int k = 0; k < 3; ++k) {
        p[(j * 3 + k) * 2] = (j == k) ? 1.f : 0.f;
        p[(j * 3 + k) * 2 + 1] = 0.f;
      }
#pragma unroll
    for (int r = 41; r < KPAD; ++r) {               // K padding rows = 0
      float* q = T + (size_t)r * NCOLS + s * 18;
#pragma unroll
      for (int e = 0; e < 18; ++e) q[e] = 0.f;
    }
  }
}

// ---------------------------------------------------------------------------
// Kernel 2: WMMA GEMM  M[48, NCOLS] = Aw[48,44] x Tw2[44, NCOLS]
// One wave per 16x16 C tile; 11 k-steps of v_wmma_f32_16x16x4_f32.
// Operand layouts per CDNA5 ISA 7.12.2:
//   A 16x4 : lanes 0-15 -> {K0,K1}, lanes 16-31 -> {K2,K3}, row = lane&15
//   B 4x16 : mirrored (rows in VGPR/lane-half), col = lane&15
//   C 16x16: VGPR p -> row p (lanes 0-15) / row p+8 (lanes 16-31)
// Fully uniform control flow (stores are unconditional into the 48-row
// padded M buffer) -> EXEC all 1s for the entire wave.
// ---------------------------------------------------------------------------
__global__ void lconv_stageB(const float* __restrict__ Aw, const float* __restrict__ T,
                             float* __restrict__ Mx) {
  int wave = blockIdx.x * (blockDim.x >> 5) + (threadIdx.x >> 5);
  int lane = threadIdx.x & 31;
  int mtile = wave / NTILES;          // 0..2
  int ntile = wave % NTILES;          // 0..18431
  int col   = ntile * 16 + (lane & 15);
  int arow  = mtile * 16 + (lane & 15);
  int khalf = (lane < 16) ? 0 : 2;

  v8f c = {};
#pragma unroll
  for (int kk = 0; kk < 11; ++kk) {
    int k0 = kk * 4 + khalf;
    v2f a, b;
    a.x = Aw[arow * KPAD + k0];
    a.y = Aw[arow * KPAD + k0 + 1];
    b.x = T[(size_t)k0 * NCOLS + col];
    b.y = T[(size_t)(k0 + 1) * NCOLS + col];
    c = __builtin_amdgcn_wmma_f32_16x16x4_f32(
        /*neg_a=*/false, a, /*neg_b=*/false, b,
        /*c_mod=*/(short)0, c, /*reuse_a=*/false, /*reuse_b=*/false);
  }

  int rbase = mtile * 16 + ((lane < 16) ? 0 : 8);
#pragma unroll
  for (int p = 0; p < 8; ++p) {
    Mx[(size_t)(rbase + p) * NCOLS + col] = c[p];   // rows >= 36 are dead pad
  }
}

// ---------------------------------------------------------------------------
// Kernel 3: per (site, u): out_w[u] = sum_{v=0..8} w2[v] @ M[u,v]
//   w2[v]   = w[v]            (v<4)
//   w2[4+v] = w[v]^dag        (v<4)
//   w2[8]   = I
// ---------------------------------------------------------------------------
__global__ void lconv_stageC(const float* __restrict__ x, const float* __restrict__ Mx,
                             float* __restrict__ out) {
  int tid = blockIdx.x * blockDim.x + threadIdx.x;   // 0 .. 4*NSITE-1
  int s = tid >> 2;
  int u = tid & 3;
  if (s >= NSITE) return;
  const float* xs = x + (size_t)s * SITE_STRIDE;

  C33 w[4];
#pragma unroll
  for (int v = 0; v < 4; ++v) loadC33(xs + (4 + v) * 18, w[v]);

  float accr[9], acci[9];
#pragma unroll
  for (int e = 0; e < 9; ++e) { accr[e] = 0.f; acci[e] = 0.f; }

#pragma unroll
  for (int v = 0; v < 9; ++v) {
    C33 M; loadC33(Mx + (size_t)(u * 9 + v) * NCOLS + s * 18, M);
    if (v < 4) {
      const C33& A = w[v];
#pragma unroll
      for (int i = 0; i < 3; ++i)
#pragma unroll
        for (int k = 0; k < 3; ++k) {
          float r = accr[i * 3 + k], m = acci[i * 3 + k];
#pragma unroll
          for (int j = 0; j < 3; ++j) {
            float ar = A.re[i * 3 + j], ai = A.im[i * 3 + j];
            float mr = M.re[j * 3 + k], mi = M.im[j * 3 + k];
            r = fmaf(ar, mr, fmaf(-ai, mi, r));
            m = fmaf(ar, mi, fmaf(ai, mr, m));
          }
          accr[i * 3 + k] = r; acci[i * 3 + k] = m;
        }
    } else if (v < 8) {
      const C33& A = w[v - 4];   // use dagger: (i,j) <- conj(A[j,i])
#pragma unroll
      for (int i = 0; i < 3; ++i)
#pragma unroll
        for (int k = 0; k < 3; ++k) {
          float r = accr[i * 3 + k], m = acci[i * 3 + k];
#pragma unroll
          for (int j = 0; j < 3; ++j) {
            float ar = A.re[j * 3 + i], ai = -A.im[j * 3 + i];
            float mr = M.re[j * 3 + k], mi = M.im[j * 3 + k];
            r = fmaf(ar, mr, fmaf(-ai, mi, r));
            m = fmaf(ar, mi, fmaf(ai, mr, m));
          }
          accr[i * 3 + k] = r; acci[i * 3 + k] = m;
        }
    } else {                      // identity term
#pragma unroll
      for (int e = 0; e < 9; ++e) { accr[e] += M.re[e]; acci[e] += M.im[e]; }
    }
  }

  float* op = out + (size_t)s * SITE_STRIDE + (4 + u) * 18;
#pragma unroll
  for (int e = 0; e < 9; ++e) { op[2 * e] = accr[e]; op[2 * e + 1] = acci[e]; }
}

// ---------------------------------------------------------------------------
extern "C" void kernel_launch(void* const* d_in, const int* in_sizes, int n_in,
                              void* d_out, int out_size, void* d_ws, size_t ws_size,
                              hipStream_t stream) {
  (void)in_sizes; (void)n_in; (void)out_size; (void)ws_size;
  const float* x      = (const float*)d_in[0];   // (1, V, 8, 3, 3, 2) f32
  const float* weight = (const float*)d_in[1];   // (4, 9, 41) f32
  float* out = (float*)d_out;                    // (1, V, 8, 3, 3, 2) f32

  float* ws = (float*)d_ws;
  float* Aw = ws;                                // 48*44 floats
  float* T  = Aw + (size_t)MPAD * KPAD;          // 44 * NCOLS floats (t_w2)
  float* Mx = T + (size_t)KPAD * NCOLS;          // 48 * NCOLS floats (M, padded)

  lconv_initAw<<<(MPAD * KPAD + 255) / 256, 256, 0, stream>>>(weight, Aw);
  lconv_stageA<<<(NSITE * 4) / 256, 256, 0, stream>>>(x, T, out);
  lconv_stageB<<<(3 * NTILES) / 8, 256, 0, stream>>>(Aw, T, Mx);   // 8 waves/block
  lconv_stageC<<<(NSITE * 4) / 256, 256, 0, stream>>>(x, Mx, out);
}